// GNN_3DModel_88072599372074
// MI455X (gfx1250) — compile-verified
//
#include <hip/hip_runtime.h>

#define D   256
#define D2  512
#define KC  32          // K-chunk staged in LDS per block

typedef float v2f __attribute__((ext_vector_type(2)));
typedef float v8f __attribute__((ext_vector_type(8)));

struct Tables { const float* t[9]; };
struct ConvP  { const float *W1, *b1, *W2, *b2, *lng, *lnb; };

// ---------------------------------------------------------------------------
// out[row, :] = sum_k tables[k][ idx[row*nt + k], : ]
__global__ void int_embed_kernel(Tables tb, int nt, const int* __restrict__ idx,
                                 float* __restrict__ out) {
  const int row = blockIdx.x;
  const int d   = threadIdx.x;
  float acc = 0.f;
  for (int k = 0; k < nt; ++k) {
    const int j = idx[row * nt + k];
    acc += tb.t[k][(long)j * D + d];
  }
  out[(long)row * D + d] = acc;
}

// out[row,:] (+)= exp(-gamma*(x[row]-c)^2) @ w + b ; centers c = step*i
__global__ void rbf_embed_kernel(const float* __restrict__ x, const float* __restrict__ w,
                                 const float* __restrict__ b, float* __restrict__ out,
                                 int nc, float step, float gamma, int accumulate) {
  __shared__ float r[64];
  const int row = blockIdx.x;
  const int d   = threadIdx.x;
  const float xv = x[row];
  if (d < nc) {
    const float t = xv - step * (float)d;
    r[d] = __expf(-gamma * t * t);
  }
  __syncthreads();
  float acc = b[d];
  for (int c = 0; c < nc; ++c) acc += r[c] * w[c * D + d];
  float* o = out + (long)row * D + d;
  if (accumulate) *o += acc; else *o = acc;
}

// ---------------------------------------------------------------------------
__global__ void copy4_kernel(const float* __restrict__ src, float* __restrict__ dst, long n4) {
  const long t = (long)blockIdx.x * blockDim.x + threadIdx.x;
  if (t < n4) ((float4*)dst)[t] = ((const float4*)src)[t];
}

__global__ void zero_kernel(float* __restrict__ p, long n) {
  const long t = (long)blockIdx.x * blockDim.x + threadIdx.x;
  if (t < n) p[t] = 0.f;
}

// h[dst] += relu(x[src] + e)   (h pre-initialized to x)
__global__ void scatter_msg_kernel(const float* __restrict__ x, const float* __restrict__ e,
                                   const int* __restrict__ src, const int* __restrict__ dst,
                                   float* __restrict__ h, long E) {
  const long t = (long)blockIdx.x * blockDim.x + threadIdx.x;
  if (t >= E * (D / 4)) return;
  const int ei = (int)(t >> 6);
  const int c  = ((int)t & 63) << 2;
  const int s  = src[ei];
  const int dd = dst[ei];
  const float4 xv = *(const float4*)(x + (long)s  * D + c);
  const float4 ev = *(const float4*)(e + (long)ei * D + c);
  float4 m;
  m.x = fmaxf(xv.x + ev.x, 0.f);
  m.y = fmaxf(xv.y + ev.y, 0.f);
  m.z = fmaxf(xv.z + ev.z, 0.f);
  m.w = fmaxf(xv.w + ev.w, 0.f);
  float* hp = h + (long)dd * D + c;
  atomicAdd(hp + 0, m.x);
  atomicAdd(hp + 1, m.y);
  atomicAdd(hp + 2, m.z);
  atomicAdd(hp + 3, m.w);
}

// ---------------------------------------------------------------------------
// C[N,M] = act(A[N,K] @ B[K,M] + bias), full f32 via V_WMMA_F32_16X16X4_F32.
//
// Block = 64 threads (2 waves), block tile = 64 rows x 64 cols.
// Each wave owns two 16-row tiles (8 v8f accumulators); one A fragment feeds
// 4 WMMAs, one B fragment 2. B is staged through LDS in pair-major layout [K/2][col][2] so the
// per-lane {K,K+1} pair is a single ds_load_b64; staging from global is
// coalesced and shared by both waves.
//
// A frag (16x4 f32): lane<16 -> row=lane, {K0,K1}; lane>=16 -> {K2,K3}.
// C/D (16x16 f32): VGPR r: lane<16 -> M=r, N=lane ; lane>=16 -> M=8+r, N=lane-16.
__global__ __launch_bounds__(64)
void gemm_bias_act_kernel(const float* __restrict__ A, const float* __restrict__ B,
                          const float* __restrict__ bias, float* __restrict__ C,
                          int K, int M, int relu) {
  __shared__ float lb[KC * 64];               // 8 KB: [KC/2][64 cols][2]
  const int row0 = blockIdx.y << 6;
  const int col0 = blockIdx.x << 6;
  const int tid  = threadIdx.x;               // 0..63
  const int wv   = tid >> 5;                  // wave 0..1
  const int lane = tid & 31;
  const int n    = lane & 15;
  const int kk   = (lane >> 4) << 1;          // 0 or 2

  v8f acc[8];
  #pragma unroll
  for (int t = 0; t < 8; ++t)
    #pragma unroll
    for (int r = 0; r < 8; ++r) acc[t][r] = 0.0f;

  // A pointers for this wave's two 16-row tiles
  const float* a0 = A + (long)(row0 + (wv << 5) + n) * K + kk;
  const float* a1 = a0 + (long)16 * K;

  // staging: thread handles one column across all KC rows of the chunk
  const float* bg = B + col0 + tid;

  for (int kg = 0; kg < K; kg += KC) {
    #pragma unroll
    for (int kr = 0; kr < KC; ++kr) {
      lb[(kr >> 1) * 128 + tid * 2 + (kr & 1)] = bg[(long)(kg + kr) * M];
    }
    __syncthreads();

    #pragma unroll
    for (int k = 0; k < KC; k += 4) {
      v2f av0, av1;
      av0.x = a0[kg + k];
      av0.y = a0[kg + k + 1];
      av1.x = a1[kg + k];
      av1.y = a1[kg + k + 1];

      const float* lp = &lb[((k + kk) >> 1) * 128 + n * 2];
      const v2f b0 = *(const v2f*)(lp + 0);
      const v2f b1 = *(const v2f*)(lp + 32);
      const v2f b2 = *(const v2f*)(lp + 64);
      const v2f b3 = *(const v2f*)(lp + 96);

      acc[0] = __builtin_amdgcn_wmma_f32_16x16x4_f32(false, av0, false, b0, (short)0, acc[0], false, false);
      acc[1] = __builtin_amdgcn_wmma_f32_16x16x4_f32(false, av0, false, b1, (short)0, acc[1], false, false);
      acc[2] = __builtin_amdgcn_wmma_f32_16x16x4_f32(false, av0, false, b2, (short)0, acc[2], false, false);
      acc[3] = __builtin_amdgcn_wmma_f32_16x16x4_f32(false, av0, false, b3, (short)0, acc[3], false, false);
      acc[4] = __builtin_amdgcn_wmma_f32_16x16x4_f32(false, av1, false, b0, (short)0, acc[4], false, false);
      acc[5] = __builtin_amdgcn_wmma_f32_16x16x4_f32(false, av1, false, b1, (short)0, acc[5], false, false);
      acc[6] = __builtin_amdgcn_wmma_f32_16x16x4_f32(false, av1, false, b2, (short)0, acc[6], false, false);
      acc[7] = __builtin_amdgcn_wmma_f32_16x16x4_f32(false, av1, false, b3, (short)0, acc[7], false, false);
    }
    __syncthreads();
  }

  const int rb = (lane >> 4) << 3;            // 0 or 8
  #pragma unroll
  for (int h = 0; h < 2; ++h) {
    const int rbase = row0 + (wv << 5) + (h << 4) + rb;
    #pragma unroll
    for (int t = 0; t < 4; ++t) {
      const int col = col0 + t * 16 + n;
      const float bv = bias[col];
      #pragma unroll
      for (int r = 0; r < 8; ++r) {
        float v = acc[h * 4 + t][r] + bv;
        if (relu) v = fmaxf(v, 0.f);
        C[(long)(rbase + r) * M + col] = v;
      }
    }
  }
}

// ---------------------------------------------------------------------------
// x[row,:] += opt_relu( LN(hin[row,:]) * g + b )      (one wave32 per row)
__global__ void ln_residual_kernel(const float* __restrict__ hin, const float* __restrict__ g,
                                   const float* __restrict__ b, float* __restrict__ x,
                                   int relu) {
  const int row  = blockIdx.x * 8 + (threadIdx.x >> 5);
  const int lane = threadIdx.x & 31;
  const float* hp = hin + (long)row * D;
  float v[8];
  float s = 0.f;
  #pragma unroll
  for (int i = 0; i < 8; ++i) { v[i] = hp[lane + 32 * i]; s += v[i]; }
  #pragma unroll
  for (int off = 16; off > 0; off >>= 1) s += __shfl_xor(s, off, 32);
  const float mean = s * (1.0f / 256.0f);
  float q = 0.f;
  #pragma unroll
  for (int i = 0; i < 8; ++i) { const float d = v[i] - mean; q += d * d; }
  #pragma unroll
  for (int off = 16; off > 0; off >>= 1) q += __shfl_xor(q, off, 32);
  const float rs = rsqrtf(q * (1.0f / 256.0f) + 1e-5f);
  float* xp = x + (long)row * D;
  #pragma unroll
  for (int i = 0; i < 8; ++i) {
    const int c = lane + 32 * i;
    float o = (v[i] - mean) * rs * g[c] + b[c];
    if (relu) o = fmaxf(o, 0.f);
    xp[c] += o;
  }
}

// ---------------------------------------------------------------------------
__global__ void pool_add_kernel(const float* __restrict__ nh, const int* __restrict__ batch,
                                float* __restrict__ sums, float* __restrict__ cnt) {
  const int row = blockIdx.x;
  const int d   = threadIdx.x;
  const int gph = batch[row];
  atomicAdd(&sums[(long)gph * D + d], nh[(long)row * D + d]);
  if (d == 0) atomicAdd(&cnt[gph], 1.0f);
}

__global__ void pool_div_kernel(const float* __restrict__ sums, const float* __restrict__ cnt,
                                float* __restrict__ graph) {
  const int gph = blockIdx.x;
  const int d   = threadIdx.x;
  graph[(long)gph * D + d] = sums[(long)gph * D + d] / fmaxf(cnt[gph], 1.0f);
}

// ---------------------------------------------------------------------------
static void run_conv(float* x, const float* e, const int* ei, int N, int E,
                     const ConvP& c, int last_act,
                     float* buf_h, float* buf_mid, float* buf_o, hipStream_t s) {
  const long n4 = (long)N * (D / 4);
  copy4_kernel<<<(unsigned)((n4 + 255) / 256), 256, 0, s>>>(x, buf_h, n4);
  const long t = (long)E * (D / 4);
  scatter_msg_kernel<<<(unsigned)((t + 255) / 256), 256, 0, s>>>(x, e, ei, ei + E, buf_h, (long)E);
  gemm_bias_act_kernel<<<dim3(D2 / 64, N / 64), 64, 0, s>>>(buf_h,   c.W1, c.b1, buf_mid, D,  D2, 1);
  gemm_bias_act_kernel<<<dim3(D  / 64, N / 64), 64, 0, s>>>(buf_mid, c.W2, c.b2, buf_o,   D2, D,  0);
  ln_residual_kernel<<<N / 8, 256, 0, s>>>(buf_o, c.lng, c.lnb, x, last_act);
}

extern "C" void kernel_launch(void* const* d_in, const int* in_sizes, int n_in,
                              void* d_out, int out_size, void* d_ws, size_t ws_size,
                              hipStream_t stream) {
  (void)in_sizes; (void)out_size; (void)ws_size;

  // ---- unpack params (insertion-order recursive flatten of setup_inputs) ----
  int p = 0;
  auto F = [&]() -> const float* { return (const float*)d_in[p++]; };

  Tables atom_tb; for (int i = 0; i < 9; ++i) atom_tb.t[i] = F();
  Tables bond_tb0; for (int i = 0; i < 3; ++i) bond_tb0.t[i] = F();
  for (int i = 3; i < 9; ++i) bond_tb0.t[i] = nullptr;
  const float* atom_rbf_w = F(); const float* atom_rbf_b = F();
  const float* bond_rbf_w = F(); const float* bond_rbf_b = F();

  struct Layer {
    Tables bemb;
    const float *brw, *brb, *arw, *arb, *nrw, *nrb;
    ConvP ab, ba, nb;
  } Ls[8];
  auto conv = [&]() -> ConvP {
    ConvP c; c.W1 = F(); c.b1 = F(); c.W2 = F(); c.b2 = F(); c.lng = F(); c.lnb = F();
    return c;
  };
  for (int l = 0; l < 8; ++l) {
    for (int i = 0; i < 3; ++i) Ls[l].bemb.t[i] = F();
    for (int i = 3; i < 9; ++i) Ls[l].bemb.t[i] = nullptr;
    Ls[l].brw = F(); Ls[l].brb = F();
    Ls[l].arw = F(); Ls[l].arb = F();
    Ls[l].nrw = F(); Ls[l].nrb = F();
    Ls[l].ab = conv(); Ls[l].ba = conv(); Ls[l].nb = conv();
  }

  // ---- data inputs: last 12 entries, in setup_inputs() order ----
  const int q = n_in - 12;
  const int*   x_atom_int          = (const int*)  d_in[q + 0];
  const float* x_atom_float        = (const float*)d_in[q + 1];
  const int*   edge_attr_atom_int  = (const int*)  d_in[q + 2];
  const float* edge_attr_atom_float= (const float*)d_in[q + 3];
  const int*   edge_index_atom     = (const int*)  d_in[q + 4];
  const int*   edge_index_nb       = (const int*)  d_in[q + 5];
  const float* edge_attr_nb        = (const float*)d_in[q + 6];
  const int*   x_bond_int          = (const int*)  d_in[q + 7];
  const float* x_bond_float        = (const float*)d_in[q + 8];
  const int*   edge_index_bond     = (const int*)  d_in[q + 9];
  const float* edge_attr_bond      = (const float*)d_in[q + 10];
  const int*   x_atom_batch        = (const int*)  d_in[q + 11];

  const int NA = 40000, NE = 160000, NNB = 320000, NB = 160000, NANG = 320000, G = 512;

  // ---- workspace layout (floats); buf_mid aliases e_buf (dead by GEMM time) ----
  float* w       = (float*)d_ws;
  float* node_h  = w;                                   // NA*D    = 10,240,000
  float* edge_h  = node_h + (long)NA * D;               // NE*D    = 40,960,000
  float* e_buf   = edge_h + (long)NE * D;               // NNB*D   = 81,920,000
  float* buf_h   = e_buf  + (long)NNB * D;              // NE*D    = 40,960,000
  float* buf_o   = buf_h  + (long)NE * D;               // NE*D    = 40,960,000
  float* psum    = buf_o  + (long)NE * D;               // G*D     = 131,072
  float* pcnt    = psum   + (long)G * D;                // G       = 512
  float* buf_mid = e_buf;                               // NE*D2   = 81,920,000 (alias)

  // ---- initial embeddings ----
  int_embed_kernel<<<NA, D, 0, stream>>>(atom_tb, 9, x_atom_int, node_h);
  rbf_embed_kernel<<<NA, D, 0, stream>>>(x_atom_float, atom_rbf_w, atom_rbf_b, node_h,
                                         20, 0.1f, 10.0f, 1);
  int_embed_kernel<<<NE, D, 0, stream>>>(bond_tb0, 3, edge_attr_atom_int, edge_h);
  rbf_embed_kernel<<<NE, D, 0, stream>>>(edge_attr_atom_float, bond_rbf_w, bond_rbf_b, edge_h,
                                         20, 0.1f, 10.0f, 1);

  // ---- 8 layers ----
  for (int l = 0; l < 8; ++l) {
    const int la = (l != 7);
    // atom-bond conv
    run_conv(node_h, edge_h, edge_index_atom, NA, NE, Ls[l].ab, la,
             buf_h, buf_mid, buf_o, stream);
    // neighbor conv
    rbf_embed_kernel<<<NNB, D, 0, stream>>>(edge_attr_nb, Ls[l].nrw, Ls[l].nrb, e_buf,
                                            50, 0.2f, 1.0f, 0);
    run_conv(node_h, e_buf, edge_index_nb, NA, NNB, Ls[l].nb, la,
             buf_h, buf_mid, buf_o, stream);
    // bond-angle conv (cur_edge overwrites edge_h; ang into e_buf)
    int_embed_kernel<<<NB, D, 0, stream>>>(Ls[l].bemb, 3, x_bond_int, edge_h);
    rbf_embed_kernel<<<NB, D, 0, stream>>>(x_bond_float, Ls[l].brw, Ls[l].brb, edge_h,
                                           20, 0.1f, 10.0f, 1);
    rbf_embed_kernel<<<NANG, D, 0, stream>>>(edge_attr_bond, Ls[l].arw, Ls[l].arb, e_buf,
                                             32, 0.1f, 10.0f, 0);
    run_conv(edge_h, e_buf, edge_index_bond, NB, NANG, Ls[l].ba, la,
             buf_h, buf_mid, buf_o, stream);
  }

  // ---- mean pooling ----
  const long zn = (long)G * D + G;
  zero_kernel<<<(unsigned)((zn + 255) / 256), 256, 0, stream>>>(psum, zn);
  pool_add_kernel<<<NA, D, 0, stream>>>(node_h, x_atom_batch, psum, pcnt);

  float* out = (float*)d_out;
  pool_div_kernel<<<G, D, 0, stream>>>(psum, pcnt, out + (long)(NA + NE) * D);

  // ---- copy node_h / edge_h to output ----
  hipMemcpyAsync(out, node_h, (size_t)NA * D * sizeof(float),
                 hipMemcpyDeviceToDevice, stream);
  hipMemcpyAsync(out + (long)NA * D, edge_h, (size_t)NE * D * sizeof(float),
                 hipMemcpyDeviceToDevice, stream);
}